// VectorizedEntropyGate_7980049236446
// MI455X (gfx1250) — compile-verified
//
#include <hip/hip_runtime.h>

// VectorizedEntropyGate for MI455X (gfx1250, wave32).
// x[b,l,d] = z[b, d%32, 2l + d/32]; h = tanh(x @ W1[l]^T + b1[l]);
// out[b,l] = entropy[b,l] * sigmoid(h . W2[l] + b2[l]).
// Block = 16 batches x 16 groups (32-col slice). z slice streamed HBM->LDS by
// the Tensor Data Mover (2D 32x32 tile iterated over 16 batches, padded LDS
// stride 1026 for bank spreading). GEMM on V_WMMA_F32_16X16X4_F32.
// ~98 KB LDS -> 3 blocks/WGP so TDM DMA overlaps other blocks' compute.

typedef __attribute__((ext_vector_type(2))) float        v2f;
typedef __attribute__((ext_vector_type(8))) float        v8f;
typedef __attribute__((ext_vector_type(4))) unsigned int v4u;
typedef __attribute__((ext_vector_type(4))) int          v4i;
typedef __attribute__((ext_vector_type(8))) int          v8i;

#define TSTRIDE 1026   // 1024 floats per batch slice + 2 pad (8B align, banks)

__global__ __launch_bounds__(256)
void entropy_gate_kernel(const float* __restrict__ z,
                         const float* __restrict__ ent,
                         const float* __restrict__ W1,
                         const float* __restrict__ b1,
                         const float* __restrict__ W2,
                         const float* __restrict__ b2,
                         float* __restrict__ out)
{
    __shared__ float tile[16 * TSTRIDE];        // 65664 B, TDM destination
    __shared__ v2f   bpack[16 * 16 * 16];       // [l'][s][entry], 32768 B

    const int tid = threadIdx.x;
    const int b0  = (blockIdx.x >> 1) * 16;     // batch tile
    const int l0  = (blockIdx.x & 1) * 16;      // group half (cols 2*l0..+31)

    // ---- Wave 0: TDM load of z[b0:b0+16, :, 2*l0 : 2*l0+32] ----
    // Per iteration: 2D tile 32 cols x 32 rows (row stride 64 floats) stored
    // contiguously (1024 floats); 16 iterations advance global by one batch
    // (2048 floats) and LDS by 1026 floats (padded stride).
    if (tid < 32) {
        const float* gsrc = z + (size_t)b0 * 2048 + 2 * l0;
        unsigned long long ga = (unsigned long long)(uintptr_t)gsrc;
        unsigned int lds      = (unsigned int)(uintptr_t)&tile[0];
        v4u g0 = { 1u,                                    // count=1 descriptor
                   lds,                                   // lds_addr (bytes)
                   (unsigned int)ga,                      // global_addr[31:0]
                   (unsigned int)(ga >> 32) | (2u << 30) };  // addr[56:32]|type=2
        v8i g1 = { (int)((2u << 16) | (1u << 19)),       // data_size=4B, iterate
                   (int)(32u << 16),                     // tensor_dim0 = 32
                   (int)(32u << 16),                     // tensor_dim1 = 32
                   (int)(32u << 16),                     // tile_dim0  = 32
                   32,                                   // tile_dim1  = 32
                   64,                                   // tensor_dim0_stride
                   0, 0 };
        v4i g2 = { 1,                                    // tensor_dim2 (unused)
                   TSTRIDE,                              // lds_addr_increment
                   2048,                                 // global_addr_increment
                   (int)(15u << 16) };                   // iterate_count: 16 iters
        v4i g3 = { 0, 0, 0, 0 };
        v8i g4 = { 0, 0, 0, 0, 0, 0, 0, 0 };             // VADDR4 unused
        __builtin_amdgcn_tensor_load_to_lds(g0, g1, g2, g3, g4, 0);
    }

    // ---- Overlap with DMA: pack W1[l0:l0+16] into B-fragment order ----
    // local idx = l'*512 + n*64 + d ; d<32 -> slot v0 (s=d/2, half=d&1)
    //                                 d>=32 -> slot v1 (s=(d-32)/2, half=(d-32)&1)
    // entry = half*8 + n (valid columns only; dead lanes mirror at read time)
    const float4* w14 = (const float4*)(W1 + l0 * 512);
    for (int i = tid; i < 2048; i += 256) {              // 8192 floats / 4
        float4 w = w14[i];
        int base = i << 2;
        #pragma unroll
        for (int c = 0; c < 4; ++c) {
            int idx = base + c;
            int lp  = idx >> 9;
            int n   = (idx >> 6) & 7;
            int d   = idx & 63;
            int dd  = d & 31;
            int s   = dd >> 1;
            int h   = dd & 1;
            float* slot = (float*)&bpack[(lp * 16 + s) * 16 + h * 8 + n];
            slot[d >> 5] = ((const float*)&w)[c];
        }
    }

    if (tid < 32) __builtin_amdgcn_s_wait_tensorcnt(0);  // wave 0: DMA done
    __syncthreads();                                     // publish tile + bpack

    const int lane = tid & 31;
    const int wave = tid >> 5;
    const int half = lane >> 4;      // 0 -> K slots {0,1}, 1 -> K slots {2,3}
    const int bidx = lane & 15;      // A-matrix row M = batch within tile
    const int ncol = lane & 15;      // B/C/D column N = hidden unit h
    const bool nv  = ncol < 8;       // only H=8 of 16 columns are real
    const int  nc  = ncol & 7;       // clamped / mirrored index

    for (int g = 0; g < 2; ++g) {
        const int lp = wave * 2 + g;             // 8 waves x 2 groups = 16
        const int l  = l0 + lp;
        v8f acc = {};

        // A fragment: tile[bidx][row 2s+half][local cols 2lp, 2lp+1]
        const int  abase = bidx * TSTRIDE + half * 32 + 2 * lp;
        // B fragment: compact pack; dead lanes mirror a valid entry (bcast)
        const v2f* bp    = &bpack[lp * 256 + half * 8 + nc];

        #pragma unroll
        for (int s = 0; s < 16; ++s) {
            v2f a  = *(const v2f*)&tile[abase + 64 * s];    // ds_load_b64
            v2f bf = bp[s * 16];                            // ds_load_b64
            // D(16x16,f32) += A(16x4,f32) x B(4x16,f32)
            acc = __builtin_amdgcn_wmma_f32_16x16x4_f32(
                false, a, false, bf, (short)0, acc, false, false);
        }

        // ---- Epilogue: tanh, dot with W2 over h (8 lanes), sigmoid ----
        float b1l = b1[l * 8 + nc];              // unconditional, clamped idx
        float w2l = W2[l * 8 + nc];
        float b1v = nv ? b1l : 0.0f;             // v_cndmask, no exec branch
        float w2v = nv ? w2l : 0.0f;
        float b2v = b2[l];

        float gsel = 0.0f;
        #pragma unroll
        for (int v = 0; v < 8; ++v) {
            // lane holds C row M = v + 8*half, column N = ncol
            float t = tanhf(acc[v] + b1v) * w2v;     // zero on dead lanes
            t += __shfl_xor(t, 1, 32);               // butterfly within octet
            t += __shfl_xor(t, 2, 32);
            t += __shfl_xor(t, 4, 32);
            if ((lane & 15) == v) gsel = t;          // rotate writer lane
        }

        if (nv) {                                    // lanes 0-7 and 16-23
            int m   = (lane & 7) + 8 * half;         // batch row in tile
            int idx = (b0 + m) * 32 + l;
            float gate = 1.0f / (1.0f + __expf(-(gsel + b2v)));
            out[idx] = ent[idx] * gate;
        }
    }
}

extern "C" void kernel_launch(void* const* d_in, const int* in_sizes, int n_in,
                              void* d_out, int out_size, void* d_ws, size_t ws_size,
                              hipStream_t stream) {
    const float* z  = (const float*)d_in[0];
    const float* e  = (const float*)d_in[1];
    const float* W1 = (const float*)d_in[2];
    const float* b1 = (const float*)d_in[3];
    const float* W2 = (const float*)d_in[4];
    const float* b2 = (const float*)d_in[5];
    float* out = (float*)d_out;

    int nB     = in_sizes[0] / 2048;   // B = elements(z) / (L*D)
    int blocks = (nB / 16) * 2;        // 16-batch x 16-group blocks: 2048
    entropy_gate_kernel<<<blocks, 256, 0, stream>>>(z, e, W1, b1, W2, b2, out);
}